// GCN_51496657879500
// MI455X (gfx1250) — compile-verified
//
#include <hip/hip_runtime.h>
#include <hip/hip_bf16.h>
#include <math.h>

typedef __attribute__((ext_vector_type(2))) float v2f;
typedef __attribute__((ext_vector_type(8))) float v8f;

#define N_NODES  50000
#define N_FEAT   128
#define N_EDGES  800000
#define N_GRAPHS 256

// ---------------------------------------------------------------------------
// OUT[M x 128] = X[M x 128] @ W[128 x 128]   (no bias; bias added post-scatter)
// One block = 16 rows x 128 cols. 8 waves, each wave owns one 16x16 N-tile.
// Uses V_WMMA_F32_16X16X4_F32 (full fp32), K-loop of 32 steps.
// A-frag (16x4): lanes 0-15 -> M=lane, K={k,k+1}; lanes 16-31 -> K={k+2,k+3}.
// C/D (16x16 f32): VGPR r -> M=r (lanes 0-15) / M=r+8 (lanes 16-31), N=lane&15.
// ---------------------------------------------------------------------------
__global__ __launch_bounds__(256) void gemm128_wmma(const float* __restrict__ X,
                                                    const float* __restrict__ W,
                                                    float* __restrict__ OUT) {
    __shared__ float sX[16 * N_FEAT];
    const int tid   = threadIdx.x;
    const int wave  = tid >> 5;
    const int lane  = tid & 31;
    const int m0    = blockIdx.x * 16;

    // Stage the 16x128 X tile into LDS (coalesced float4 loads).
    const float4* sv = (const float4*)(X + (size_t)m0 * N_FEAT);
    float4* dv = (float4*)sX;
    #pragma unroll
    for (int i = tid; i < (16 * N_FEAT) / 4; i += 256) dv[i] = sv[i];
    __syncthreads();

    const int n0    = wave * 16;
    const int mrow  = lane & 15;
    const int khalf = lane >> 4;      // 0: K pair {k,k+1}; 1: {k+2,k+3}
    v8f acc = {};

    #pragma unroll 8
    for (int k = 0; k < N_FEAT; k += 4) {
        const int ka = k + 2 * khalf;
        v2f a, b;
        a.x = sX[mrow * N_FEAT + ka];
        a.y = sX[mrow * N_FEAT + ka + 1];
        b.x = W[(size_t)ka       * N_FEAT + n0 + mrow];
        b.y = W[(size_t)(ka + 1) * N_FEAT + n0 + mrow];
        acc = __builtin_amdgcn_wmma_f32_16x16x4_f32(
            /*neg_a=*/false, a, /*neg_b=*/false, b,
            /*c_mod=*/(short)0, acc, /*reuse_a=*/false, /*reuse_b=*/false);
    }

    const int nc    = n0 + mrow;
    const int rbase = khalf * 8;
    #pragma unroll
    for (int r = 0; r < 8; r++) {
        OUT[(size_t)(m0 + rbase + r) * N_FEAT + nc] = acc[r];
    }
}

// ---------------------------------------------------------------------------
// Degree accumulation + d^-1/2 conversion (computed once, reused by all convs)
// ---------------------------------------------------------------------------
__global__ void deg_accum(const int* __restrict__ dst, float* __restrict__ deg, int nE) {
    int i = blockIdx.x * blockDim.x + threadIdx.x;
    if (i < nE) atomicAdd(&deg[dst[i]], 1.0f);
}

__global__ void deg_to_dis(float* __restrict__ deg, int n) {
    int i = blockIdx.x * blockDim.x + threadIdx.x;
    if (i < n) {
        float d = deg[i];
        deg[i] = (d > 0.0f) ? rsqrtf(fmaxf(d, 1.0f)) : 0.0f;
    }
}

// ---------------------------------------------------------------------------
// Edge scatter: one wave per edge; lane handles a float4 chunk of the 128 feats.
// out[dst] += h[src] * dis[src]*dis[dst]  via global f32 atomics.
// Grid = N_EDGES/8 blocks of 256 threads (8 waves) — exact cover.
// ---------------------------------------------------------------------------
__global__ __launch_bounds__(256) void edge_scatter(const float* __restrict__ h,
                                                    const int* __restrict__ src,
                                                    const int* __restrict__ dst,
                                                    const float* __restrict__ dis,
                                                    float* __restrict__ out) {
    const int e    = blockIdx.x * 8 + (threadIdx.x >> 5);
    const int lane = threadIdx.x & 31;
    const int s = src[e];
    const int d = dst[e];
    const float nrm = dis[s] * dis[d];
    const float4 hv = ((const float4*)(h + (size_t)s * N_FEAT))[lane];
    float* o = out + (size_t)d * N_FEAT + lane * 4;
    atomicAdd(o + 0, hv.x * nrm);
    atomicAdd(o + 1, hv.y * nrm);
    atomicAdd(o + 2, hv.z * nrm);
    atomicAdd(o + 3, hv.w * nrm);
}

// ---------------------------------------------------------------------------
// h[i] = (h[i] + b[i%128]), optional ReLU
// ---------------------------------------------------------------------------
__global__ void bias_act(float* __restrict__ h, const float* __restrict__ b,
                         int relu, int n) {
    int i = blockIdx.x * blockDim.x + threadIdx.x;
    if (i < n) {
        float v = h[i] + b[i & (N_FEAT - 1)];
        if (relu) v = fmaxf(v, 0.0f);
        h[i] = v;
    }
}

// ---------------------------------------------------------------------------
// Mean-pool: atomic feature sums per graph + node counts per graph
// ---------------------------------------------------------------------------
__global__ void pool_sum(const float* __restrict__ h, const int* __restrict__ batch,
                         float* __restrict__ sums, int n) {
    int i = blockIdx.x * blockDim.x + threadIdx.x;
    if (i < n) {
        int node = i >> 7;           // /128
        int c    = i & (N_FEAT - 1);
        atomicAdd(&sums[(size_t)batch[node] * N_FEAT + c], h[i]);
    }
}

__global__ void pool_cnt(const int* __restrict__ batch, float* __restrict__ cnts, int n) {
    int i = blockIdx.x * blockDim.x + threadIdx.x;
    if (i < n) atomicAdd(&cnts[batch[i]], 1.0f);
}

// ---------------------------------------------------------------------------
// Head: g = relu(mean @ Wf1 + bf1) [64]; logits = g @ Wl + bl [2]; softmax.
// One block per graph, 64 threads (one per hidden unit).
// ---------------------------------------------------------------------------
__global__ __launch_bounds__(64) void head(const float* __restrict__ sums,
                                           const float* __restrict__ cnts,
                                           const float* __restrict__ Wf1,
                                           const float* __restrict__ bf1,
                                           const float* __restrict__ Wl,
                                           const float* __restrict__ bl,
                                           float* __restrict__ out) {
    __shared__ float sh[64];
    const int g = blockIdx.x;
    const int t = threadIdx.x;
    const float inv = 1.0f / fmaxf(cnts[g], 1.0f);
    float acc = bf1[t];
    #pragma unroll 4
    for (int k = 0; k < N_FEAT; k++)
        acc += (sums[(size_t)g * N_FEAT + k] * inv) * Wf1[(size_t)k * 64 + t];
    sh[t] = fmaxf(acc, 0.0f);
    __syncthreads();
    if (t == 0) {
        float l0 = bl[0], l1 = bl[1];
        for (int j = 0; j < 64; j++) {
            l0 += sh[j] * Wl[j * 2 + 0];
            l1 += sh[j] * Wl[j * 2 + 1];
        }
        float m  = fmaxf(l0, l1);
        float e0 = __expf(l0 - m), e1 = __expf(l1 - m);
        float s  = e0 + e1;
        out[g * 2 + 0] = e0 / s;
        out[g * 2 + 1] = e1 / s;
    }
}

// ---------------------------------------------------------------------------
extern "C" void kernel_launch(void* const* d_in, const int* in_sizes, int n_in,
                              void* d_out, int out_size, void* d_ws, size_t ws_size,
                              hipStream_t stream) {
    const float* x     = (const float*)d_in[0];
    const int*   ei    = (const int*)  d_in[1];   // [2, N_EDGES]
    const int*   batch = (const int*)  d_in[2];
    const float* W1    = (const float*)d_in[3];
    const float* b1    = (const float*)d_in[4];
    const float* W2    = (const float*)d_in[5];
    const float* b2    = (const float*)d_in[6];
    const float* Wf1   = (const float*)d_in[7];
    const float* bf1   = (const float*)d_in[8];
    const float* Wl    = (const float*)d_in[9];
    const float* bl    = (const float*)d_in[10];
    const int* srcI = ei;
    const int* dstI = ei + N_EDGES;

    const size_t NH = (size_t)N_NODES * N_FEAT;
    float* bufA = (float*)d_ws;                       // 25.6 MB
    float* bufB = bufA + NH;                          // 25.6 MB
    float* dis  = bufB + NH;                          // 200 KB
    float* sums = dis + N_NODES;                      // 128 KB
    float* cnts = sums + (size_t)N_GRAPHS * N_FEAT;   // 1 KB

    const int ETHREADS = 256;
    const int NHB  = (int)((NH + 255) / 256);
    const int GEMB = N_NODES / 16;                    // 3125, exact
    const int SCB  = N_EDGES / 8;                     // 100000, exact

    // normalizers (once)
    hipMemsetAsync(dis, 0, N_NODES * sizeof(float), stream);
    deg_accum<<<(N_EDGES + ETHREADS - 1) / ETHREADS, ETHREADS, 0, stream>>>(dstI, dis, N_EDGES);
    deg_to_dis<<<(N_NODES + ETHREADS - 1) / ETHREADS, ETHREADS, 0, stream>>>(dis, N_NODES);

    // conv1: relu(scatter(x@W1) + b1)
    gemm128_wmma<<<GEMB, 256, 0, stream>>>(x, W1, bufA);
    hipMemsetAsync(bufB, 0, NH * sizeof(float), stream);
    edge_scatter<<<SCB, 256, 0, stream>>>(bufA, srcI, dstI, dis, bufB);
    bias_act<<<NHB, 256, 0, stream>>>(bufB, b1, 1, (int)NH);

    // conv2: relu(scatter(h1@W2) + b2)
    gemm128_wmma<<<GEMB, 256, 0, stream>>>(bufB, W2, bufA);
    hipMemsetAsync(bufB, 0, NH * sizeof(float), stream);
    edge_scatter<<<SCB, 256, 0, stream>>>(bufA, srcI, dstI, dis, bufB);
    bias_act<<<NHB, 256, 0, stream>>>(bufB, b2, 1, (int)NH);

    // conv3: scatter(h2@W2) + b2 (no relu)
    gemm128_wmma<<<GEMB, 256, 0, stream>>>(bufB, W2, bufA);
    hipMemsetAsync(bufB, 0, NH * sizeof(float), stream);
    edge_scatter<<<SCB, 256, 0, stream>>>(bufA, srcI, dstI, dis, bufB);
    bias_act<<<NHB, 256, 0, stream>>>(bufB, b2, 0, (int)NH);

    // mean-pool + MLP head + softmax
    hipMemsetAsync(sums, 0, ((size_t)N_GRAPHS * N_FEAT + N_GRAPHS) * sizeof(float), stream);
    pool_sum<<<NHB, 256, 0, stream>>>(bufB, batch, sums, (int)NH);
    pool_cnt<<<(N_NODES + ETHREADS - 1) / ETHREADS, ETHREADS, 0, stream>>>(batch, cnts, N_NODES);
    head<<<N_GRAPHS, 64, 0, stream>>>(sums, cnts, Wf1, bf1, Wl, bl, (float*)d_out);
}